// NodeBlockV2_27444841021659
// MI455X (gfx1250) — compile-verified
//
#include <hip/hip_runtime.h>

// ---------------------------------------------------------------------------
// EquiformerV2-style node block on gfx1250 (CDNA5, wave32).
// All edge-shared-weight GEMMs use V_WMMA_F32_16X16X4_F32 (exact fp32).
// Workspace requirement: ~663.2 MB.
// ---------------------------------------------------------------------------

typedef float v2f __attribute__((ext_vector_type(2)));
typedef float v8f __attribute__((ext_vector_type(8)));

#define EDGES 64000
#define NODES 10000

__device__ __forceinline__ v8f wmma_f32(v2f a, v2f b, v8f c) {
  return __builtin_amdgcn_wmma_f32_16x16x4_f32(false, a, false, b, (short)0, c, false, false);
}

__device__ __forceinline__ int idx0_coeff(int i) { return (i == 0) ? 0 : ((i == 1) ? 2 : 6); }

__device__ __forceinline__ float sigmoidf_(float x) { return 1.f / (1.f + expf(-x)); }
__device__ __forceinline__ float siluf_(float x) { return x * sigmoidf_(x); }

// order-preserving float->uint key for atomicMax
__device__ __forceinline__ unsigned fkey(float x) {
  unsigned u = __float_as_uint(x);
  return (u >> 31) ? ~u : (u | 0x80000000u);
}
__device__ __forceinline__ float funkey(unsigned k) {
  unsigned u = (k & 0x80000000u) ? (k ^ 0x80000000u) : ~k;
  return __uint_as_float(u);
}

// ---------------------------------------------------------------------------
__global__ void k_clear(float* __restrict__ p, long long n) {
  long long i = (long long)blockIdx.x * blockDim.x + threadIdx.x;
  long long stride = (long long)gridDim.x * blockDim.x;
  for (; i < n; i += stride) p[i] = 0.f;
}

// ---------------------------------------------------------------------------
// Radial MLP: 256 -> 64 (LN,silu) -> 64 (LN,silu) -> 576.  One block per edge.
__global__ void k_radial(const float* __restrict__ edist, const float* __restrict__ semb,
                         const float* __restrict__ temb, const int* __restrict__ anum,
                         const int* __restrict__ eidx,
                         const float* __restrict__ w1, const float* __restrict__ b1,
                         const float* __restrict__ g1, const float* __restrict__ bb1,
                         const float* __restrict__ w2, const float* __restrict__ b2,
                         const float* __restrict__ g2, const float* __restrict__ bb2,
                         const float* __restrict__ w3, const float* __restrict__ b3,
                         float* __restrict__ rad) {
  __shared__ float xe[256];
  __shared__ float hb[64];
  const int e = blockIdx.x, t = threadIdx.x;
  const int src = eidx[e], tgt = eidx[EDGES + e];
  const int as = anum[src], at = anum[tgt];
  for (int i = t; i < 256; i += 64) {
    float v;
    if (i < 128) v = edist[(size_t)e * 128 + i];
    else if (i < 192) v = semb[(size_t)as * 64 + (i - 128)];
    else v = temb[(size_t)at * 64 + (i - 192)];
    xe[i] = v;
  }
  __syncthreads();
  float h = b1[t];
  for (int i = 0; i < 256; ++i) h += xe[i] * w1[i * 64 + t];
  hb[t] = h;
  __syncthreads();
  float s = 0.f, s2 = 0.f;
  for (int i = 0; i < 64; ++i) { float v = hb[i]; s += v; s2 += v * v; }
  float mu = s * (1.f / 64.f);
  float var = s2 * (1.f / 64.f) - mu * mu;
  float vv = (h - mu) * rsqrtf(var + 1e-5f) * g1[t] + bb1[t];
  vv = siluf_(vv);
  __syncthreads(); hb[t] = vv; __syncthreads();
  float h2 = b2[t];
  for (int i = 0; i < 64; ++i) h2 += hb[i] * w2[i * 64 + t];
  __syncthreads(); hb[t] = h2; __syncthreads();
  s = 0.f; s2 = 0.f;
  for (int i = 0; i < 64; ++i) { float v = hb[i]; s += v; s2 += v * v; }
  mu = s * (1.f / 64.f);
  var = s2 * (1.f / 64.f) - mu * mu;
  vv = (h2 - mu) * rsqrtf(var + 1e-5f) * g2[t] + bb2[t];
  vv = siluf_(vv);
  __syncthreads(); hb[t] = vv; __syncthreads();
  float* ro = rad + (size_t)e * 576;
  for (int j = 0; j < 9; ++j) {
    const int col = j * 64 + t;
    float r = b3[col];
    for (int i = 0; i < 64; ++i) r += hb[i] * w3[i * 576 + col];
    ro[col] = r;
  }
}

// ---------------------------------------------------------------------------
// msgrot[e,i,c] = sum_j wigner[e,i,j] * cat(x[src], x[tgt], edge_fea)[j,c]
__global__ void k_rotate(const float* __restrict__ x, const float* __restrict__ efea,
                         const float* __restrict__ wig, const int* __restrict__ eidx,
                         float* __restrict__ msgrot) {
  __shared__ float cat[9 * 96];
  __shared__ float wg[81];
  const int e = blockIdx.x, t = threadIdx.x;  // 96 threads
  const int src = eidx[e], tgt = eidx[EDGES + e];
  if (t < 81) wg[t] = wig[(size_t)e * 81 + t];
  for (int j = 0; j < 9; ++j) {
    float v;
    if (t < 32) v = x[(size_t)src * 288 + j * 32 + t];
    else if (t < 64) v = x[(size_t)tgt * 288 + j * 32 + (t - 32)];
    else v = efea[(size_t)e * 288 + j * 32 + (t - 64)];
    cat[j * 96 + t] = v;
  }
  __syncthreads();
  float* mo = msgrot + (size_t)e * 864;
  for (int i = 0; i < 9; ++i) {
    float s = 0.f;
#pragma unroll
    for (int j = 0; j < 9; ++j) s += wg[i * 9 + j] * cat[j * 96 + t];
    mo[i * 96 + t] = s;
  }
}

// ---------------------------------------------------------------------------
// conv1 m=0 path: [E x 288] @ [288 x 576] + bias. cols<384 -> extra, else out1[{0,2,6}]
__global__ void k_gemm_c1_x0(const float* __restrict__ msgrot, const float* __restrict__ rad,
                             const float* __restrict__ W, const float* __restrict__ bias,
                             float* __restrict__ extra, float* __restrict__ out1) {
  const int NT = (EDGES / 16) * (576 / 16);
  const int wid = blockIdx.x * (blockDim.x >> 5) + (threadIdx.x >> 5);
  if (wid >= NT) return;
  const int tn = wid % 36, tm = wid / 36;
  const int lane = threadIdx.x & 31, half = lane >> 4, lr = lane & 15;
  const int m0 = tm * 16, n0 = tn * 16;
  const int erow = m0 + lr;
  const int col = n0 + lr;
  const float* radp = rad + (size_t)erow * 576;
  const float* rotp = msgrot + (size_t)erow * 864;
  v8f acc = {0.f, 0.f, 0.f, 0.f, 0.f, 0.f, 0.f, 0.f};
  for (int k0 = 0; k0 < 288; k0 += 4) {
    const int ks = k0 + half * 2;
    v2f a, b;
#pragma unroll
    for (int t = 0; t < 2; ++t) {
      const int k = ks + t;
      const int i = k / 96, c = k - i * 96;
      const float av = rotp[idx0_coeff(i) * 96 + c] * radp[k];
      if (t == 0) a.x = av; else a.y = av;
    }
    b.x = W[(size_t)ks * 576 + col];
    b.y = W[(size_t)(ks + 1) * 576 + col];
    acc = wmma_f32(a, b, acc);
  }
  const float bv = bias[col];
#pragma unroll
  for (int r = 0; r < 8; ++r) {
    const int e = m0 + r + half * 8;
    const float val = acc[r] + bv;
    if (col < 384) {
      extra[(size_t)e * 384 + col] = val;
    } else {
      const int cc = col - 384;
      out1[(size_t)e * 576 + idx0_coeff(cc >> 6) * 64 + (cc & 63)] = val;
    }
  }
}

// ---------------------------------------------------------------------------
// conv1 m>0 paths: rows (e, r) of [2E x KD] @ {Wr,Wi} [KD x ND]; combine neg/pos.
// MC==0: m=1 (coeffs neg{1,5} pos{3,7}, rad off 288), MC==1: m=2 (neg{4} pos{8}, rad off 480)
template <int KD, int ND, int MC>
__global__ void k_gemm_c1_m(const float* __restrict__ msgrot, const float* __restrict__ rad,
                            const float* __restrict__ Wr, const float* __restrict__ Wi,
                            float* __restrict__ out1) {
  const int NT = (2 * EDGES / 16) * (ND / 16);
  const int wid = blockIdx.x * (blockDim.x >> 5) + (threadIdx.x >> 5);
  if (wid >= NT) return;
  const int ntn = ND / 16;
  const int tn = wid % ntn, tm = wid / ntn;
  const int lane = threadIdx.x & 31, half = lane >> 4, lr = lane & 15;
  const int m0 = tm * 16, n0 = tn * 16;
  const int row = m0 + lr;
  const int e = row >> 1, rr = row & 1;
  const int col = n0 + lr;
  const float* rotp = msgrot + (size_t)e * 864;
  const float* radp = rad + (size_t)e * 576 + ((MC == 0) ? 288 : 480);
  v8f aR = {0.f, 0.f, 0.f, 0.f, 0.f, 0.f, 0.f, 0.f};
  v8f aI = {0.f, 0.f, 0.f, 0.f, 0.f, 0.f, 0.f, 0.f};
  for (int k0 = 0; k0 < KD; k0 += 4) {
    const int ks = k0 + half * 2;
    v2f a, br, bi;
#pragma unroll
    for (int t = 0; t < 2; ++t) {
      const int k = ks + t;
      float av;
      if (MC == 0) {
        const int kk = k / 96, c = k - kk * 96;
        const int coeff = rr ? (kk ? 7 : 3) : (kk ? 5 : 1);
        av = rotp[coeff * 96 + c] * radp[k];
      } else {
        const int coeff = rr ? 8 : 4;
        av = rotp[coeff * 96 + k] * radp[k];
      }
      if (t == 0) a.x = av; else a.y = av;
    }
    br.x = Wr[(size_t)ks * ND + col]; br.y = Wr[(size_t)(ks + 1) * ND + col];
    bi.x = Wi[(size_t)ks * ND + col]; bi.y = Wi[(size_t)(ks + 1) * ND + col];
    aR = wmma_f32(a, br, aR);
    aI = wmma_f32(a, bi, aI);
  }
  int cneg, cpos, c;
  if (MC == 0) { const int j = col >> 6; c = col & 63; cneg = j ? 5 : 1; cpos = j ? 7 : 3; }
  else { c = col; cneg = 4; cpos = 8; }
#pragma unroll
  for (int p = 0; p < 4; ++p) {
    const int rowe = m0 + 2 * p + half * 8;  // even row -> rr==0
    const int ee = rowe >> 1;
    out1[(size_t)ee * 576 + cneg * 64 + c] = aR[2 * p] - aI[2 * p + 1];
    out1[(size_t)ee * 576 + cpos * 64 + c] = aR[2 * p + 1] + aI[2 * p];
  }
}

// ---------------------------------------------------------------------------
// gating: coeff 0 -> silu, coeffs 1..8 -> * sigmoid(gate) (gate row 0 for l=1, 1 for l=2)
__global__ void k_gate(float* __restrict__ out1, const float* __restrict__ extra) {
  const long long idx = (long long)blockIdx.x * 256 + threadIdx.x;
  if (idx >= (long long)EDGES * 576) return;
  const int e = (int)(idx / 576);
  const int rem = (int)(idx - (long long)e * 576);
  const int kc = rem >> 6, c = rem & 63;
  float v = out1[idx];
  if (kc == 0) {
    v = siluf_(v);
  } else {
    const int lrep = (kc < 4) ? 0 : 1;
    v *= sigmoidf_(extra[(size_t)e * 384 + 256 + lrep * 64 + c]);
  }
  out1[idx] = v;
}

// attention logits: LN(32) -> smooth_leaky_relu -> dot alpha_dot
__global__ void k_logits(const float* __restrict__ extra, const float* __restrict__ ang,
                         const float* __restrict__ anb, const float* __restrict__ adot,
                         float* __restrict__ abuf) {
  const int idx = blockIdx.x * 256 + threadIdx.x;
  if (idx >= EDGES * 8) return;
  const int e = idx >> 3, h = idx & 7;
  const float* xp = extra + (size_t)e * 384 + h * 32;
  float s = 0.f, s2 = 0.f;
  for (int k = 0; k < 32; ++k) { float v = xp[k]; s += v; s2 += v * v; }
  const float mu = s * (1.f / 32.f);
  const float var = s2 * (1.f / 32.f) - mu * mu;
  const float inv = rsqrtf(var + 1e-5f);
  float acc = 0.f;
  for (int k = 0; k < 32; ++k) {
    float v = (xp[k] - mu) * inv * ang[k] + anb[k];
    const float sl = 0.6f * v + 0.4f * v * (2.f * sigmoidf_(v) - 1.f);
    acc += sl * adot[h * 32 + k];
  }
  abuf[idx] = acc;
}

// ---------------------------------------------------------------------------
// conv2 m=0 path: [E x 192] @ [192 x 384] + bias -> out2[{0,2,6}]
__global__ void k_gemm_c2_x0(const float* __restrict__ g1, const float* __restrict__ W,
                             const float* __restrict__ bias, float* __restrict__ out2) {
  const int NT = (EDGES / 16) * (384 / 16);
  const int wid = blockIdx.x * (blockDim.x >> 5) + (threadIdx.x >> 5);
  if (wid >= NT) return;
  const int tn = wid % 24, tm = wid / 24;
  const int lane = threadIdx.x & 31, half = lane >> 4, lr = lane & 15;
  const int m0 = tm * 16, n0 = tn * 16;
  const int erow = m0 + lr, col = n0 + lr;
  const float* ap = g1 + (size_t)erow * 576;
  v8f acc = {0.f, 0.f, 0.f, 0.f, 0.f, 0.f, 0.f, 0.f};
  for (int k0 = 0; k0 < 192; k0 += 4) {
    const int ks = k0 + half * 2;
    v2f a, b;
#pragma unroll
    for (int t = 0; t < 2; ++t) {
      const int k = ks + t;
      const float av = ap[idx0_coeff(k >> 6) * 64 + (k & 63)];
      if (t == 0) a.x = av; else a.y = av;
    }
    b.x = W[(size_t)ks * 384 + col];
    b.y = W[(size_t)(ks + 1) * 384 + col];
    acc = wmma_f32(a, b, acc);
  }
  const float bv = bias[col];
  const int coeff = idx0_coeff(col >> 7);
  const int c = col & 127;
#pragma unroll
  for (int r = 0; r < 8; ++r) {
    const int e = m0 + r + half * 8;
    out2[(size_t)e * 1152 + coeff * 128 + c] = acc[r] + bv;
  }
}

// conv2 m>0 paths (no rad); MC==0: m=1 coeffs {1,5}/{3,7}; MC==1: m=2 {4}/{8}
template <int KD, int ND, int MC>
__global__ void k_gemm_c2_m(const float* __restrict__ g1, const float* __restrict__ Wr,
                            const float* __restrict__ Wi, float* __restrict__ out2) {
  const int NT = (2 * EDGES / 16) * (ND / 16);
  const int wid = blockIdx.x * (blockDim.x >> 5) + (threadIdx.x >> 5);
  if (wid >= NT) return;
  const int ntn = ND / 16;
  const int tn = wid % ntn, tm = wid / ntn;
  const int lane = threadIdx.x & 31, half = lane >> 4, lr = lane & 15;
  const int m0 = tm * 16, n0 = tn * 16;
  const int row = m0 + lr;
  const int e = row >> 1, rr = row & 1;
  const int col = n0 + lr;
  const float* ap = g1 + (size_t)e * 576;
  v8f aR = {0.f, 0.f, 0.f, 0.f, 0.f, 0.f, 0.f, 0.f};
  v8f aI = {0.f, 0.f, 0.f, 0.f, 0.f, 0.f, 0.f, 0.f};
  for (int k0 = 0; k0 < KD; k0 += 4) {
    const int ks = k0 + half * 2;
    v2f a, br, bi;
#pragma unroll
    for (int t = 0; t < 2; ++t) {
      const int k = ks + t;
      float av;
      if (MC == 0) {
        const int kk = k >> 6, c = k & 63;
        const int coeff = rr ? (kk ? 7 : 3) : (kk ? 5 : 1);
        av = ap[coeff * 64 + c];
      } else {
        av = ap[(rr ? 8 : 4) * 64 + k];
      }
      if (t == 0) a.x = av; else a.y = av;
    }
    br.x = Wr[(size_t)ks * ND + col]; br.y = Wr[(size_t)(ks + 1) * ND + col];
    bi.x = Wi[(size_t)ks * ND + col]; bi.y = Wi[(size_t)(ks + 1) * ND + col];
    aR = wmma_f32(a, br, aR);
    aI = wmma_f32(a, bi, aI);
  }
  int cneg, cpos, c;
  if (MC == 0) { const int j = col >> 7; c = col & 127; cneg = j ? 5 : 1; cpos = j ? 7 : 3; }
  else { c = col; cneg = 4; cpos = 8; }
#pragma unroll
  for (int p = 0; p < 4; ++p) {
    const int rowe = m0 + 2 * p + half * 8;
    const int ee = rowe >> 1;
    out2[(size_t)ee * 1152 + cneg * 128 + c] = aR[2 * p] - aI[2 * p + 1];
    out2[(size_t)ee * 1152 + cpos * 128 + c] = aR[2 * p + 1] + aI[2 * p];
  }
}

// ---------------------------------------------------------------------------
__global__ void k_attmax(const float* __restrict__ abuf, const int* __restrict__ eidx,
                         unsigned* __restrict__ amaxkey) {
  const int idx = blockIdx.x * 256 + threadIdx.x;
  if (idx >= EDGES * 8) return;
  const int e = idx >> 3, h = idx & 7;
  const int tgt = eidx[EDGES + e];
  atomicMax(&amaxkey[tgt * 8 + h], fkey(abuf[idx]));
}

__global__ void k_attexp(float* __restrict__ abuf, const int* __restrict__ eidx,
                         const unsigned* __restrict__ amaxkey, float* __restrict__ denom) {
  const int idx = blockIdx.x * 256 + threadIdx.x;
  if (idx >= EDGES * 8) return;
  const int e = idx >> 3, h = idx & 7;
  const int tgt = eidx[EDGES + e];
  const float ex = expf(abuf[idx] - funkey(amaxkey[tgt * 8 + h]));
  abuf[idx] = ex;
  atomicAdd(&denom[tgt * 8 + h], ex);
}

// alpha-weight, transpose-rotate, segment-sum into node accumulator
__global__ void k_scatter(const float* __restrict__ out2, const float* __restrict__ abuf,
                          const float* __restrict__ denom, const float* __restrict__ wig,
                          const int* __restrict__ eidx, float* __restrict__ node) {
  __shared__ float wg[81];
  __shared__ float v[1152];
  __shared__ float al[8];
  const int e = blockIdx.x, t = threadIdx.x;  // 128 threads
  const int tgt = eidx[EDGES + e];
  if (t < 81) wg[t] = wig[(size_t)e * 81 + t];
  if (t < 8) al[t] = abuf[e * 8 + t] / (denom[tgt * 8 + t] + 1e-16f);
  __syncthreads();
  for (int idx = t; idx < 1152; idx += 128) {
    const int c = idx & 127;
    v[idx] = out2[(size_t)e * 1152 + idx] * al[c >> 4];
  }
  __syncthreads();
  const int c = t;  // 0..127
  for (int i = 0; i < 9; ++i) {
    float s = 0.f;
#pragma unroll
    for (int j = 0; j < 9; ++j) s += wg[j * 9 + i] * v[j * 128 + c];
    atomicAdd(&node[(size_t)tgt * 1152 + i * 128 + c], s);
  }
}

// ---------------------------------------------------------------------------
// out[n,k,:] = node[n,k,:] @ proj_w[PROJ_L[k]] (+ proj_b at k==0)
__global__ void k_proj(const float* __restrict__ node, const float* __restrict__ projw,
                       const float* __restrict__ projb, float* __restrict__ out) {
  const int NT = 9 * (NODES / 16) * 2;  // 11250
  const int wid = blockIdx.x * (blockDim.x >> 5) + (threadIdx.x >> 5);
  if (wid >= NT) return;
  const int kcoef = wid / 1250;
  const int rest = wid % 1250;
  const int tm = rest >> 1, tn = rest & 1;
  const int l = (kcoef == 0) ? 0 : ((kcoef < 4) ? 1 : 2);
  const float* W = projw + (size_t)l * 128 * 32;
  const int lane = threadIdx.x & 31, half = lane >> 4, lr = lane & 15;
  const int m0 = tm * 16, n0 = tn * 16;
  const int nrow = m0 + lr, col = n0 + lr;
  const float* ap = node + (size_t)nrow * 1152 + kcoef * 128;
  v8f acc = {0.f, 0.f, 0.f, 0.f, 0.f, 0.f, 0.f, 0.f};
  for (int k0 = 0; k0 < 128; k0 += 4) {
    const int ks = k0 + half * 2;
    v2f a, b;
    a.x = ap[ks]; a.y = ap[ks + 1];
    b.x = W[ks * 32 + col]; b.y = W[(ks + 1) * 32 + col];
    acc = wmma_f32(a, b, acc);
  }
  const float bv = (kcoef == 0) ? projb[col] : 0.f;
#pragma unroll
  for (int r = 0; r < 8; ++r) {
    const int n = m0 + r + half * 8;
    out[(size_t)n * 288 + kcoef * 32 + col] = acc[r] + bv;
  }
}

// ---------------------------------------------------------------------------
extern "C" void kernel_launch(void* const* d_in, const int* in_sizes, int n_in,
                              void* d_out, int out_size, void* d_ws, size_t ws_size,
                              hipStream_t stream) {
  (void)in_sizes; (void)n_in; (void)out_size; (void)ws_size;
  const float* x     = (const float*)d_in[0];
  const float* efea  = (const float*)d_in[1];
  const float* edist = (const float*)d_in[2];
  const float* wig   = (const float*)d_in[3];
  const float* semb  = (const float*)d_in[4];
  const float* temb  = (const float*)d_in[5];
  const float* rw1 = (const float*)d_in[6];  const float* rb1 = (const float*)d_in[7];
  const float* rg1 = (const float*)d_in[8];  const float* rbb1 = (const float*)d_in[9];
  const float* rw2 = (const float*)d_in[10]; const float* rb2 = (const float*)d_in[11];
  const float* rg2 = (const float*)d_in[12]; const float* rbb2 = (const float*)d_in[13];
  const float* rw3 = (const float*)d_in[14]; const float* rb3 = (const float*)d_in[15];
  const float* c1w0 = (const float*)d_in[16]; const float* c1b0 = (const float*)d_in[17];
  const float* c1wr1 = (const float*)d_in[18]; const float* c1wi1 = (const float*)d_in[19];
  const float* c1wr2 = (const float*)d_in[20]; const float* c1wi2 = (const float*)d_in[21];
  const float* c2w0 = (const float*)d_in[22]; const float* c2b0 = (const float*)d_in[23];
  const float* c2wr1 = (const float*)d_in[24]; const float* c2wi1 = (const float*)d_in[25];
  const float* c2wr2 = (const float*)d_in[26]; const float* c2wi2 = (const float*)d_in[27];
  const float* ang = (const float*)d_in[28]; const float* anb = (const float*)d_in[29];
  const float* adot = (const float*)d_in[30];
  const float* projw = (const float*)d_in[31]; const float* projb = (const float*)d_in[32];
  const int* anum = (const int*)d_in[33];
  const int* eidx = (const int*)d_in[34];

  char* ws = (char*)d_ws;
  // region A (reused): rad + msgrot live during conv1; out2 overlays them for conv2+
  float*    rad     = (float*)(ws + 0ull);               // E*576*4   = 147,456,000
  float*    msgrot  = (float*)(ws + 147456000ull);       // E*864*4   = 221,184,000
  float*    out2    = (float*)(ws + 0ull);               // E*1152*4  = 294,912,000 (reuse)
  float*    out1    = (float*)(ws + 368640000ull);       // E*576*4   = 147,456,000
  float*    extra   = (float*)(ws + 516096000ull);       // E*384*4   =  98,304,000
  float*    abuf    = (float*)(ws + 614400000ull);       // E*8*4     =   2,048,000
  unsigned* amaxkey = (unsigned*)(ws + 616448000ull);    // N*8*4     =     320,000
  float*    denom   = (float*)(ws + 616768000ull);       // N*8*4     =     320,000
  float*    node    = (float*)(ws + 617088000ull);       // N*1152*4  =  46,080,000
  // total workspace: 663,168,000 bytes

  // zero amaxkey + denom + node (contiguous): 11,680,000 words
  k_clear<<<2048, 256, 0, stream>>>((float*)(ws + 616448000ull), 11680000LL);

  k_radial<<<EDGES, 64, 0, stream>>>(edist, semb, temb, anum, eidx,
                                     rw1, rb1, rg1, rbb1, rw2, rb2, rg2, rbb2, rw3, rb3, rad);
  k_rotate<<<EDGES, 96, 0, stream>>>(x, efea, wig, eidx, msgrot);

  // SO(2) conv #1 (WMMA fp32)
  k_gemm_c1_x0<<<18000, 256, 0, stream>>>(msgrot, rad, c1w0, c1b0, extra, out1);
  k_gemm_c1_m<192, 128, 0><<<8000, 256, 0, stream>>>(msgrot, rad, c1wr1, c1wi1, out1);
  k_gemm_c1_m<96, 64, 1><<<4000, 256, 0, stream>>>(msgrot, rad, c1wr2, c1wi2, out1);

  k_gate<<<(EDGES * 576) / 256, 256, 0, stream>>>(out1, extra);
  k_logits<<<(EDGES * 8) / 256, 256, 0, stream>>>(extra, ang, anb, adot, abuf);

  // SO(2) conv #2 (WMMA fp32)
  k_gemm_c2_x0<<<12000, 256, 0, stream>>>(out1, c2w0, c2b0, out2);
  k_gemm_c2_m<128, 256, 0><<<16000, 256, 0, stream>>>(out1, c2wr1, c2wi1, out2);
  k_gemm_c2_m<64, 128, 1><<<8000, 256, 0, stream>>>(out1, c2wr2, c2wi2, out2);

  // segment softmax over tgt
  k_attmax<<<(EDGES * 8) / 256, 256, 0, stream>>>(abuf, eidx, amaxkey);
  k_attexp<<<(EDGES * 8) / 256, 256, 0, stream>>>(abuf, eidx, amaxkey, denom);

  // weight, rotate back, scatter-add to nodes
  k_scatter<<<EDGES, 128, 0, stream>>>(out2, abuf, denom, wig, eidx, node);

  // per-l output projection (WMMA fp32)
  k_proj<<<1407, 256, 0, stream>>>(node, projw, projb, (float*)d_out);
}